// MCB_28570122453942
// MI455X (gfx1250) — compile-verified
//
#include <hip/hip_runtime.h>
#include <math.h>

// ---------------------------------------------------------------------------
// MCB (count-sketch + circular convolution) fully fused for gfx1250 (MI455X).
//
//   N = 16000 = 128 * 125 -> four-step FFT; both small DFTs run as dense
//   128x128 fp32 matmuls on V_WMMA_F32_16X16X4_F32 (Gauss 3-mult complex).
//
//   One workgroup (512 threads = 16 wave32, 4 waves/SIMD for latency hiding)
//   per batch row; the whole row pipeline lives in 2 complex 128x128 LDS
//   buffers (~264 KB) plus ~3 KB of root-of-unity tables -> needs CDNA5's
//   320 KB/WGP LDS.  Sketch built in place with ds_add_f32 LDS atomics; HBM
//   traffic is just the compulsory ~165 MB (inputs+output), ~7 us @ 23.3 TB/s.
// ---------------------------------------------------------------------------

typedef float v2f __attribute__((ext_vector_type(2)));
typedef float v8f __attribute__((ext_vector_type(8)));

#define NFFT 16000
#define NN1  128     // fast dimension (radix-128 stage)
#define NN2  125     // slow dimension (radix-125 stage, zero padded to 128)
#define LP   129     // LDS row pitch (128 + 1 pad -> conflict-free columns)
#define NTHREADS 512 // 16 wave32 -> 4 waves per SIMD32

static __device__ __forceinline__ v8f wmma4(v2f a, v2f b, v8f c) {
  // D = A(16x4) * B(4x16) + C(16x16), fp32
  return __builtin_amdgcn_wmma_f32_16x16x4_f32(
      /*neg_a=*/false, a, /*neg_b=*/false, b,
      /*c_mod=*/(short)0, c, /*reuse_a=*/false, /*reuse_b=*/false);
}

// ---------------------------------------------------------------------------
// Step 1: T[n1][k2] = sum_{n2} M[n1][n2] * E5[n2][k2], then apply the
// four-step twiddle w_N^{n1*k2}.  E5 entries come from 125-entry LDS tables
// (cos, sin, cos+sin) with incremental mod-125 indexing -> no trans ops in
// the MAC loop.  Complex product via Gauss: 3 WMMA chains instead of 4.
// Wave w owns tile row (w>>1) and tile columns (w&1)*4 .. +3.
// ---------------------------------------------------------------------------
static __device__ __forceinline__ void step1_twiddle(
    const float* __restrict__ srR, const float* __restrict__ srI,
    float* __restrict__ dsR, float* __restrict__ dsI,
    const float* __restrict__ tC125, const float* __restrict__ tS125,
    const float* __restrict__ tCS125, int lane, int wave) {
  const float TWO_PI = 6.28318530717958647692f;
  const int laneHi = lane >> 4;   // 0: lanes 0-15, 1: lanes 16-31
  const int laneLo = lane & 15;
  const int mBase = (wave >> 1) * 16;
  const int ntBeg = (wave & 1) * 4;
  const int mr = mBase + laneLo;  // A-fragment row (n1)

  for (int nt = ntBeg; nt < ntBeg + 4; ++nt) {
    const int nBase = nt * 16;
    const int nc = nBase + laneLo;          // B-fragment col (k2)
    const bool ncv = nc < NN2;
    const int ncm = ncv ? nc : 0;           // keep indices in range when padded
    // incremental mod-125 indices for rows r0 = 4*kk + 2*laneHi, r1 = r0+1
    int t0 = (2 * laneHi * ncm) % NN2;      // row r0 at kk = 0
    const int stepRow = ncm;                // r -> r+1   (< 125)
    const int stepIter = (4 * ncm) % NN2;   // kk -> kk+1

    v8f p1 = {}, p2 = {}, p3 = {};
    for (int kk = 0; kk < 32; ++kk) {
      const int kb = kk * 4 + (laneHi << 1);  // this half-wave's K pair
      // --- A fragment (16x4) from LDS
      v2f ar, ai;
      ar.x = srR[mr * LP + kb];     ar.y = srR[mr * LP + kb + 1];
      ai.x = srI[mr * LP + kb];     ai.y = srI[mr * LP + kb + 1];
      const v2f as = ar + ai;
      // --- B fragment = E5[n2][k2] from tables (zero in the padded region)
      int t1 = t0 + stepRow; t1 -= (t1 >= NN2) ? NN2 : 0;
      const bool v0 = ncv && (kb < NN2);
      const bool v1 = ncv && ((kb + 1) < NN2);
      v2f br, bi, bs;
      br.x = v0 ? tC125[t0]  : 0.0f;  br.y = v1 ? tC125[t1]  : 0.0f;
      bi.x = v0 ? tS125[t0]  : 0.0f;  bi.y = v1 ? tS125[t1]  : 0.0f;
      bs.x = v0 ? tCS125[t0] : 0.0f;  bs.y = v1 ? tCS125[t1] : 0.0f;
      p1 = wmma4(ar, br, p1);   // Ar*Br
      p2 = wmma4(ai, bi, p2);   // Ai*Bi
      p3 = wmma4(as, bs, p3);   // (Ar+Ai)*(Br+Bi)
      t0 += stepIter; t0 -= (t0 >= NN2) ? NN2 : 0;
    }
    // --- store with twiddle w_N^{row*col}
    for (int r = 0; r < 8; ++r) {
      const int row = mBase + r + (laneHi << 3);  // n1
      const int col = nc;                         // k2
      const float tre = p1[r] - p2[r];
      const float tim = p3[r] - p1[r] - p2[r];
      int t = row * col;                 // <= 127*127 < 2*16000
      if (t >= NFFT) t -= NFFT;
      float sv, cv;
      __sincosf(-TWO_PI * (float)t * (1.0f / (float)NFFT), &sv, &cv);
      dsR[row * LP + col] = tre * cv - tim * sv;
      dsI[row * LP + col] = tre * sv + tim * cv;
    }
  }
}

// ---------------------------------------------------------------------------
// Step 3: U[k1][k2] = sum_{n1} E2[k1][n1] * T[n1][k2].  E2 entries come from
// 128-entry LDS tables with incremental mod-128 indexing.  Either stores the
// complex result to LDS, or (final inverse pass) writes real(U)/N to global.
// ---------------------------------------------------------------------------
template <bool TO_GLOBAL>
static __device__ __forceinline__ void step3(
    const float* __restrict__ srR, const float* __restrict__ srI,
    float* __restrict__ dsR, float* __restrict__ dsI,
    float* __restrict__ outRow,
    const float* __restrict__ tC128, const float* __restrict__ tS128,
    const float* __restrict__ tCS128, int lane, int wave) {
  const int laneHi = lane >> 4;
  const int laneLo = lane & 15;
  const int mBase = (wave >> 1) * 16;
  const int ntBeg = (wave & 1) * 4;
  const int k1 = mBase + laneLo;          // A-fragment row

  const int stepRow = k1 & 127;           // n1 -> n1+1
  const int stepIter = (4 * k1) & 127;    // kk -> kk+1

  for (int nt = ntBeg; nt < ntBeg + 4; ++nt) {
    const int nBase = nt * 16;
    const int nc = nBase + laneLo;        // k2
    int t0 = (k1 * (laneHi << 1)) & 127;  // n1 = 2*laneHi at kk = 0
    v8f p1 = {}, p2 = {}, p3 = {};
    for (int kk = 0; kk < 32; ++kk) {
      const int kb = kk * 4 + (laneHi << 1);
      // --- A fragment = E2[k1][n1] from tables
      const int t1 = (t0 + stepRow) & 127;
      v2f ar, ai, as;
      ar.x = tC128[t0];   ar.y = tC128[t1];
      ai.x = tS128[t0];   ai.y = tS128[t1];
      as.x = tCS128[t0];  as.y = tCS128[t1];
      // --- B fragment (4x16) from LDS: rows kb, kb+1 at column nc
      v2f br, bi;
      br.x = srR[kb * LP + nc];  br.y = srR[(kb + 1) * LP + nc];
      bi.x = srI[kb * LP + nc];  bi.y = srI[(kb + 1) * LP + nc];
      const v2f bs = br + bi;
      p1 = wmma4(ar, br, p1);
      p2 = wmma4(ai, bi, p2);
      p3 = wmma4(as, bs, p3);
      t0 = (t0 + stepIter) & 127;
    }
    for (int r = 0; r < 8; ++r) {
      const int row = mBase + r + (laneHi << 3);  // k1
      const int col = nBase + laneLo;             // k2
      if (TO_GLOBAL) {
        // inverse: y[n] = conj(V)[n]/N, n = 125*k1 + k2;  keep real part
        if (col < NN2)
          outRow[row * NN2 + col] = (p1[r] - p2[r]) * (1.0f / (float)NFFT);
      } else {
        dsR[row * LP + col] = p1[r] - p2[r];
        dsI[row * LP + col] = p3[r] - p1[r] - p2[r];
      }
    }
  }
}

// ---------------------------------------------------------------------------
// Fused kernel: one workgroup per batch row.
// ---------------------------------------------------------------------------
__global__ __launch_bounds__(NTHREADS) void mcb_fused_kernel(
    const float* __restrict__ img, const float* __restrict__ seq,
    const int* __restrict__ h, const float* __restrict__ s,
    float* __restrict__ out, int d) {
  __shared__ float aR[NN1 * LP], aI[NN1 * LP];
  __shared__ float bR[NN1 * LP], bI[NN1 * LP];
  __shared__ float tC128[128], tS128[128], tCS128[128];
  __shared__ float tC125[NN2], tS125[NN2], tCS125[NN2];

  const float TWO_PI = 6.28318530717958647692f;
  const int tid = threadIdx.x;
  const int lane = tid & 31;
  const int wave = tid >> 5;
  const int b = blockIdx.x;

  // ---- root-of-unity tables (cos, sin, cos+sin) + zero buffer A
  for (int t = tid; t < 128; t += NTHREADS) {
    float sv, cv;
    __sincosf(-TWO_PI * (float)t * (1.0f / 128.0f), &sv, &cv);
    tC128[t] = cv; tS128[t] = sv; tCS128[t] = cv + sv;
  }
  for (int t = tid; t < NN2; t += NTHREADS) {
    float sv, cv;
    __sincosf(-TWO_PI * (float)t * (1.0f / (float)NN2), &sv, &cv);
    tC125[t] = cv; tS125[t] = sv; tCS125[t] = cv + sv;
  }
  for (int i = tid; i < NN1 * LP; i += NTHREADS) { aR[i] = 0.0f; aI[i] = 0.0f; }
  __syncthreads();

  // ---- count-sketch scatter directly into LDS (ds_add_f32 atomics)
  // time index k -> matrix (n1 = k mod 128, n2 = k div 128);  z = sk_img + i*sk_seq
  const float* imgRow = img + (size_t)b * (size_t)d;
  const float* seqRow = seq + (size_t)b * (size_t)d;
  for (int i = tid; i < d; i += NTHREADS) {
    const int k = h[i];
    const float sg = s[i];
    const int row = k & 127;
    const int col = k >> 7;
    atomicAdd(&aR[row * LP + col], imgRow[i] * sg);
    atomicAdd(&aI[row * LP + col], seqRow[i] * sg);
  }
  __syncthreads();

  // ---- forward FFT of z:  A --(step1+twiddle)--> B --(step3)--> A  (= Z)
  step1_twiddle(aR, aI, bR, bI, tC125, tS125, tCS125, lane, wave);
  __syncthreads();
  step3<false>(bR, bI, aR, aI, nullptr, tC128, tS128, tCS128, lane, wave);
  __syncthreads();

  // ---- zero buffer B (pad columns 125..127 must be 0 for the inverse pass)
  for (int i = tid; i < NN1 * LP; i += NTHREADS) { bR[i] = 0.0f; bI[i] = 0.0f; }
  __syncthreads();

  // ---- split A,B spectra from Z, multiply, conjugate, relayout to time grid
  // Z[f] lives at (k1 = f/125, k2 = f%125); conj(P)[f] goes to
  // (m1 = f mod 128, m2 = f div 128) as the input of the inverse transform.
  for (int f = tid; f < NFFT; f += NTHREADS) {
    const int k1 = f / 125, k2 = f - k1 * 125;
    const float zfr = aR[k1 * LP + k2], zfi = aI[k1 * LP + k2];
    const int g = (f == 0) ? 0 : (NFFT - f);
    const int g1 = g / 125, g2 = g - g1 * 125;
    const float zgr = aR[g1 * LP + g2], zgi = aI[g1 * LP + g2];
    // A = (Z[f] + conj Z[g])/2 ;  B = (Z[f] - conj Z[g])/(2i)
    const float Are = 0.5f * (zfr + zgr), Aim = 0.5f * (zfi - zgi);
    const float Bre = 0.5f * (zfi + zgi), Bim = 0.5f * (zgr - zfr);
    const float Pr = Are * Bre - Aim * Bim;
    const float Pi = Are * Bim + Aim * Bre;
    const int m1 = f & 127, m2 = f >> 7;
    bR[m1 * LP + m2] = Pr;
    bI[m1 * LP + m2] = -Pi;   // conj -> inverse FFT via forward machinery
  }
  __syncthreads();

  // ---- inverse FFT:  B --(step1+twiddle)--> A --(step3, real/N)--> out
  step1_twiddle(bR, bI, aR, aI, tC125, tS125, tCS125, lane, wave);
  __syncthreads();
  step3<true>(aR, aI, nullptr, nullptr, out + (size_t)b * (size_t)NFFT,
              tC128, tS128, tCS128, lane, wave);
}

extern "C" void kernel_launch(void* const* d_in, const int* in_sizes, int n_in,
                              void* d_out, int out_size, void* d_ws,
                              size_t ws_size, hipStream_t stream) {
  const float* img = (const float*)d_in[0];   // [B, d] f32
  const float* seq = (const float*)d_in[1];   // [B, d] f32
  const int* h = (const int*)d_in[2];         // [d] i32 bucket ids
  const float* s = (const float*)d_in[3];     // [d] f32 +/-1 signs
  float* out = (float*)d_out;                 // [B, 16000] f32

  const int d = in_sizes[2];
  const int B = in_sizes[0] / d;

  mcb_fused_kernel<<<B, NTHREADS, 0, stream>>>(img, seq, h, s, out, d);
}